// NTMCell_79551384256782
// MI455X (gfx1250) — compile-verified
//
#include <hip/hip_runtime.h>
#include <math.h>
#include <stdint.h>

// ---- problem dims ----
#define BDIM   256
#define INDIM  256
#define OUTDIM 256
#define STDIM  512
#define HDIM   4
#define NSH    3
#define MDIM   64
#define NDIM   1024
#define KDIM   1024          // D = IN + ST + READ
#define PPH    198           // 3+1+64+64+64+1+1
#define UPD    792           // H*PPH

typedef float v2f __attribute__((ext_vector_type(2)));
typedef float v8f __attribute__((ext_vector_type(8)));

__device__ __forceinline__ float sigmoidf_(float x) { return 1.0f / (1.0f + expf(-x)); }
__device__ __forceinline__ float softplusf_(float x) { return (x > 20.0f) ? x : log1pf(expf(x)); }

// Convert a generic pointer into LDS into its 32-bit LDS byte offset
// (AS(3) pointers on AMDGPU are 32-bit offsets from the wave's LDS base).
__device__ __forceinline__ uint32_t lds_off_(const void* p) {
#if defined(__HIP_DEVICE_COMPILE__)
    return (uint32_t)(uintptr_t)(__attribute__((address_space(3))) const void*)p;
#else
    (void)p; return 0u;
#endif
}

// ============================================================================
// Kernel 1: gate f = sigmoid([state,input] @ W_i2w^T + b), gated wt -> ws,
//           fill c[:,0:768] = [input, state], and (block 0) zero the dummy row
//           used by the GEMM edge tiles.
// ============================================================================
__global__ __launch_bounds__(256) void ntm_gate_kernel(
    const float* __restrict__ tm_input, const float* __restrict__ tm_state,
    const float* __restrict__ wt, const float* __restrict__ W_i2w,
    const float* __restrict__ b_i2w,
    float* __restrict__ wtg, float* __restrict__ cbuf,
    float* __restrict__ zrow)
{
    const int b = blockIdx.x;
    const int t = threadIdx.x;
    __shared__ float fsh[HDIM];

    if (b == 0) {                           // zero dummy K-row for GEMM edge tiles
        for (int j = t; j < KDIM; j += 256) zrow[j] = 0.0f;
    }

    if (t < 128) {                          // waves 0..3: one head each
        const int h = t >> 5, lane = t & 31;
        float s = 0.0f;
        for (int j = lane; j < STDIM + INDIM; j += 32) {
            float x = (j < STDIM) ? tm_state[b * STDIM + j]
                                  : tm_input[b * INDIM + (j - STDIM)];
            s += W_i2w[h * (STDIM + INDIM) + j] * x;
        }
        #pragma unroll
        for (int off = 16; off; off >>= 1) s += __shfl_xor(s, off, 32);
        if (lane == 0) fsh[h] = sigmoidf_(s + b_i2w[h]);
    } else {                                // waves 4..7: fill c[:,0:768]
        const int t2 = t - 128;
        for (int j = t2; j < INDIM + STDIM; j += 128) {
            cbuf[(size_t)b * KDIM + j] =
                (j < INDIM) ? tm_input[b * INDIM + j]
                            : tm_state[b * STDIM + (j - INDIM)];
        }
    }
    __syncthreads();

    // wt_g = f*onehot0 + (1-f)*wt
    for (int idx = t; idx < HDIM * NDIM; idx += 256) {
        const int n = idx & (NDIM - 1);
        const float f = fsh[idx >> 10];
        const float w = wt[(size_t)b * HDIM * NDIM + idx];
        wtg[(size_t)b * HDIM * NDIM + idx] = (n == 0 ? f : 0.0f) + (1.0f - f) * w;
    }
}

// ============================================================================
// Kernel 2: read[b,h,m] = sum_n wt_g[b,h,n] * mem[b,n,m]  -> c[:,768:1024]
// One block per batch; thread = (h,m); coalesced over m; wt broadcast via L0.
// ============================================================================
__global__ __launch_bounds__(256) void ntm_read_kernel(
    const float* __restrict__ wtg, const float* __restrict__ mem,
    float* __restrict__ cbuf)
{
    const int b = blockIdx.x;
    const int t = threadIdx.x;
    const int h = t >> 6, m = t & 63;
    const float* __restrict__ wrow = wtg + (size_t)b * HDIM * NDIM + h * NDIM;
    const float* __restrict__ mcol = mem + (size_t)b * NDIM * MDIM + m;
    float acc = 0.0f;
    #pragma unroll 4
    for (int n = 0; n < NDIM; ++n)
        acc += wrow[n] * mcol[(size_t)n * MDIM];
    cbuf[(size_t)b * KDIM + (INDIM + STDIM) + t] = acc;
}

// ============================================================================
// Kernel 3: Y[b, o] = act( X[b,:] @ W[o,:] + bias[o] )  via V_WMMA_F32_16X16X4_F32
//  - X tile (16 batches x 1024 K, shared by all 4 waves) staged ONCE into LDS
//    with GLOBAL_LOAD_ASYNC_TO_LDS_B128 (ASYNCcnt), rows padded to 1028 floats
//    so the (row, kk)/(row, kk+2) fragment reads are LDS-bank-conflict free.
//  - W (A-matrix) stream is register double-buffered: next k-step's fragments
//    are in flight while the current 4 WMMAs execute.
//  - Edge feature rows read a zeroed dummy row: no divergence in the hot loop.
// ============================================================================
#define XROWS 16
#define XPAD  1028     // 1024 + 4 floats pad: bank stride 4 -> all 64 banks hit once

__global__ __launch_bounds__(128) void gemm_wmma_f32(
    const float* __restrict__ X,     // [B, KDIM]
    const float* __restrict__ W,     // [n_out, KDIM]
    const float* __restrict__ bias,  // [n_out]
    const float* __restrict__ zrow,  // [KDIM] zeros
    float* __restrict__ Y,           // [B, ldy]
    int n_out, int ldy, int act)
{
    __shared__ float lds_x[XROWS * XPAD];

    const int lane = threadIdx.x;          // 0..31
    const int wy   = threadIdx.y;          // 0..3
    const int t    = wy * 32 + lane;       // 0..127
    const int half = lane >> 4;            // K-half selector
    const int r    = lane & 15;            // row-within-16
    const int b0   = blockIdx.x * 16;
    const int o0   = (blockIdx.y * 4 + wy) * 64;

    // ---- async-stage the shared X tile into LDS (16 B per thread per op) ----
    for (int e = t * 4; e < XROWS * KDIM; e += 128 * 4) {
        const int row = e >> 10, col = e & (KDIM - 1);
        const float* gp = X + (size_t)(b0 + row) * KDIM + col;
        const uint32_t lo = lds_off_(&lds_x[row * XPAD + col]);
#if defined(__HIP_DEVICE_COMPILE__)
        asm volatile("global_load_async_to_lds_b128 %0, %1, off"
                     :: "v"(lo), "v"(gp) : "memory");
#else
        (void)gp; (void)lo;
#endif
    }
#if defined(__HIP_DEVICE_COMPILE__)
    asm volatile("s_wait_asynccnt 0x0" ::: "memory");
#endif
    __syncthreads();

    const float* arow[4];
    #pragma unroll
    for (int tt = 0; tt < 4; ++tt) {
        const int o = o0 + tt * 16 + r;
        arow[tt] = (o < n_out) ? (W + (size_t)o * KDIM) : zrow;
    }

    v8f acc[4];
    #pragma unroll
    for (int tt = 0; tt < 4; ++tt)
        #pragma unroll
        for (int i = 0; i < 8; ++i) acc[tt][i] = 0.0f;

    const float* __restrict__ xl = &lds_x[r * XPAD];   // this lane's batch row

    // ---- software-pipelined K loop: prefetch (k+4) while WMMA-ing (k) ----
    const int kk0 = 2 * half;
    v2f bf_c = *(const v2f*)(xl + kk0);
    v2f af_c[4];
    #pragma unroll
    for (int tt = 0; tt < 4; ++tt) af_c[tt] = *(const v2f*)(arow[tt] + kk0);

    for (int k = 0; k < KDIM; k += 4) {
        const int kn = (k + 4 < KDIM) ? (k + 4 + 2 * half) : (2 * half); // wrap: dummy
        v2f bf_n = *(const v2f*)(xl + kn);
        v2f af_n[4];
        #pragma unroll
        for (int tt = 0; tt < 4; ++tt) af_n[tt] = *(const v2f*)(arow[tt] + kn);

        #pragma unroll
        for (int tt = 0; tt < 4; ++tt) {
#if defined(__HIP_DEVICE_COMPILE__) && __has_builtin(__builtin_amdgcn_wmma_f32_16x16x4_f32)
            acc[tt] = __builtin_amdgcn_wmma_f32_16x16x4_f32(
                false, af_c[tt], false, bf_c, (short)0, acc[tt], false, false);
#else
            // exact cross-lane emulation of the 16x16x4 f32 WMMA fragment math
            #pragma unroll
            for (int i = 0; i < 8; ++i) {
                const int mrow = half * 8 + i;
                float s = acc[tt][i];
                #pragma unroll
                for (int k2 = 0; k2 < 4; ++k2) {
                    float av = __shfl((k2 & 1) ? af_c[tt].y : af_c[tt].x, mrow + 16 * (k2 >> 1), 32);
                    float bv = __shfl((k2 & 1) ? bf_c.y     : bf_c.x,     r    + 16 * (k2 >> 1), 32);
                    s += av * bv;
                }
                acc[tt][i] = s;
            }
#endif
        }
        bf_c = bf_n;
        #pragma unroll
        for (int tt = 0; tt < 4; ++tt) af_c[tt] = af_n[tt];
    }

    // D layout: vgpr i holds row (8*half + i), col = lane&15
    #pragma unroll
    for (int tt = 0; tt < 4; ++tt) {
        #pragma unroll
        for (int i = 0; i < 8; ++i) {
            const int o = o0 + tt * 16 + half * 8 + i;
            if (o < n_out) {
                float v = acc[tt][i] + bias[o];
                if (act) v = sigmoidf_(v);
                Y[(size_t)(b0 + r) * ldy + o] = v;
            }
        }
    }
}

// ============================================================================
// Kernel 4: fused interface activations + erase/add write + cosine sim +
//           softmax/gate/shift/sharpen -> out_mem, out_wt. One block per batch.
// ============================================================================
__global__ __launch_bounds__(256) void ntm_update_kernel(
    const float* __restrict__ wtg, const float* __restrict__ mem,
    const float* __restrict__ ud,                 // [B, UPD]
    float* __restrict__ out_mem, float* __restrict__ out_wt)
{
    const int b = blockIdx.x;
    const int t = threadIdx.x;

    __shared__ float s_er[HDIM][MDIM], s_ad[HDIM][MDIM], s_kn[HDIM][MDIM];
    __shared__ float s_ssh[HDIM][NSH], s_gamma[HDIM], s_beta[HDIM], s_g[HDIM];
    __shared__ float s_knorm[HDIM], s_hv[HDIM];
    __shared__ float s_sim[HDIM][NDIM];
    __shared__ float s_wt[HDIM][NDIM];
    __shared__ float red[HDIM][64];

    // ---- Phase A: activate interface params; thread = (h, m) ----
    const int h4 = t >> 6, m = t & 63;
    const float* __restrict__ udh = ud + (size_t)b * UPD + h4 * PPH;
    s_er[h4][m] = sigmoidf_(udh[4 + m]);
    s_ad[h4][m] = tanhf(udh[68 + m]);
    s_kn[h4][m] = tanhf(udh[132 + m]);
    if (t < HDIM) {
        const float* __restrict__ u = ud + (size_t)b * UPD + t * PPH;
        float sp0 = softplusf_(u[0]), sp1 = softplusf_(u[1]), sp2 = softplusf_(u[2]);
        float mx = fmaxf(sp0, fmaxf(sp1, sp2));
        float e0 = expf(sp0 - mx), e1 = expf(sp1 - mx), e2 = expf(sp2 - mx);
        float z = e0 + e1 + e2;
        s_ssh[t][0] = e0 / z; s_ssh[t][1] = e1 / z; s_ssh[t][2] = e2 / z;
        s_gamma[t] = 1.0f + softplusf_(u[3]);
        s_beta[t]  = softplusf_(u[196]);
        s_g[t]     = sigmoidf_(u[197]);
    }
    __syncthreads();
    if (t < HDIM) {
        float s = 0.0f;
        for (int j = 0; j < MDIM; ++j) { float v = s_kn[t][j]; s += v * v; }
        s_knorm[t] = 1.0f / (sqrtf(s) + 1e-8f);
    }
    __syncthreads();
    s_kn[h4][m] *= s_knorm[h4];
    __syncthreads();

    // ---- Phase B: single pass over mem rows: write + norm + dots ----
    const int wv = t >> 5, lane = t & 31;
    const int m0 = lane, m1 = lane + 32;
    for (int n = wv; n < NDIM; n += 8) {
        float wh[HDIM];
        #pragma unroll
        for (int h = 0; h < HDIM; ++h)
            wh[h] = wtg[(size_t)b * HDIM * NDIM + h * NDIM + n];

        const float* __restrict__ mrow = mem + ((size_t)b * NDIM + n) * MDIM;
        float v0 = mrow[m0], v1 = mrow[m1];

        float k0 = 1.0f, k1 = 1.0f, a0 = 0.0f, a1 = 0.0f;
        #pragma unroll
        for (int h = 0; h < HDIM; ++h) {
            k0 *= (1.0f - wh[h] * s_er[h][m0]);
            k1 *= (1.0f - wh[h] * s_er[h][m1]);
            a0 += wh[h] * s_ad[h][m0];
            a1 += wh[h] * s_ad[h][m1];
        }
        const float nm0 = v0 * k0 + a0;
        const float nm1 = v1 * k1 + a1;

        float* __restrict__ orow = out_mem + ((size_t)b * NDIM + n) * MDIM;
        orow[m0] = nm0; orow[m1] = nm1;

        float nrm = nm0 * nm0 + nm1 * nm1;
        float d[HDIM];
        #pragma unroll
        for (int h = 0; h < HDIM; ++h)
            d[h] = s_kn[h][m0] * nm0 + s_kn[h][m1] * nm1;

        #pragma unroll
        for (int off = 16; off; off >>= 1) {
            nrm += __shfl_xor(nrm, off, 32);
            #pragma unroll
            for (int h = 0; h < HDIM; ++h) d[h] += __shfl_xor(d[h], off, 32);
        }
        if (lane == 0) {
            const float inv = 1.0f / (sqrtf(nrm) + 1e-8f);
            #pragma unroll
            for (int h = 0; h < HDIM; ++h) s_sim[h][n] = d[h] * inv;
        }
    }
    __syncthreads();

    // ---- Phase C: softmax(beta*sim) per head, gate-interp with wt_g ----
    const int hc = t >> 6, c = t & 63;       // 64 threads per head, 16 n's each
    float mx = -3.0e38f;
    for (int j = 0; j < 16; ++j) mx = fmaxf(mx, s_sim[hc][c + 64 * j]);
    red[hc][c] = mx;
    __syncthreads();
    if (t < HDIM) {
        float mm = red[t][0];
        for (int j = 1; j < 64; ++j) mm = fmaxf(mm, red[t][j]);
        s_hv[t] = mm;
    }
    __syncthreads();
    {
        const float beta = s_beta[hc], mxh = s_hv[hc];
        float ps = 0.0f;
        for (int j = 0; j < 16; ++j) {
            const int n = c + 64 * j;
            const float e = expf(beta * (s_sim[hc][n] - mxh));
            s_sim[hc][n] = e;
            ps += e;
        }
        red[hc][c] = ps;
    }
    __syncthreads();
    if (t < HDIM) {
        float ss = 0.0f;
        for (int j = 0; j < 64; ++j) ss += red[t][j];
        s_hv[t] = 1.0f / ss;
    }
    __syncthreads();
    {
        const float zin = s_hv[hc], gh = s_g[hc];
        for (int j = 0; j < 16; ++j) {
            const int n = c + 64 * j;
            const float wc   = s_sim[hc][n] * zin;
            const float wold = wtg[(size_t)b * HDIM * NDIM + hc * NDIM + n];
            s_wt[hc][n] = gh * wc + (1.0f - gh) * wold;
        }
    }
    __syncthreads();

    // ---- Phase D: circular 3-tap shift, sharpen, L1 renorm ----
    {
        const float sh0 = s_ssh[hc][0], sh1 = s_ssh[hc][1], sh2 = s_ssh[hc][2];
        const float gam = s_gamma[hc];
        float tmp[16];
        float psum = 0.0f;
        for (int j = 0; j < 16; ++j) {
            const int n = c + 64 * j;
            // out[n] = s0*wt[n+1] + s1*wt[n] + s2*wt[n-1]  (circular)
            float v = sh0 * s_wt[hc][(n + 1) & (NDIM - 1)]
                    + sh1 * s_wt[hc][n]
                    + sh2 * s_wt[hc][(n - 1) & (NDIM - 1)];
            v = powf(v + 1e-12f, gam);
            tmp[j] = v;
            psum += v;
        }
        red[hc][c] = psum;
        __syncthreads();
        if (t < HDIM) {
            float ss = 0.0f;
            for (int j = 0; j < 64; ++j) ss += red[t][j];
            s_hv[t] = 1.0f / ss;
        }
        __syncthreads();
        const float inv = s_hv[hc];
        for (int j = 0; j < 16; ++j) {
            const int n = c + 64 * j;
            out_wt[(size_t)b * HDIM * NDIM + hc * NDIM + n] = tmp[j] * inv;
        }
    }
}

// ============================================================================
extern "C" void kernel_launch(void* const* d_in, const int* in_sizes, int n_in,
                              void* d_out, int out_size, void* d_ws, size_t ws_size,
                              hipStream_t stream)
{
    const float* tm_input = (const float*)d_in[0];
    const float* tm_state = (const float*)d_in[1];
    const float* wt       = (const float*)d_in[2];
    const float* mem      = (const float*)d_in[3];
    const float* W_i2w    = (const float*)d_in[4];
    const float* b_i2w    = (const float*)d_in[5];
    const float* W_i2s    = (const float*)d_in[6];
    const float* b_i2s    = (const float*)d_in[7];
    const float* W_i2o    = (const float*)d_in[8];
    const float* b_i2o    = (const float*)d_in[9];
    const float* W_i2u    = (const float*)d_in[10];
    const float* b_i2u    = (const float*)d_in[11];

    float* out = (float*)d_out;
    float* out_y   = out;                                   // [256,256]
    float* out_st  = out + BDIM * OUTDIM;                   // [256,512]
    float* out_wt  = out + BDIM * (OUTDIM + STDIM);         // [256,4,1024]
    float* out_mem = out + BDIM * (OUTDIM + STDIM + HDIM * NDIM); // [256,1024,64]

    float* ws   = (float*)d_ws;
    float* wtg  = ws;                                       // 256*4*1024
    float* cbuf = wtg + (size_t)BDIM * HDIM * NDIM;         // 256*1024
    float* udb  = cbuf + (size_t)BDIM * KDIM;               // 256*792
    float* zrow = udb + (size_t)BDIM * UPD;                 // 1024 zeros

    ntm_gate_kernel<<<BDIM, 256, 0, stream>>>(tm_input, tm_state, wt, W_i2w, b_i2w, wtg, cbuf, zrow);
    ntm_read_kernel<<<BDIM, 256, 0, stream>>>(wtg, mem, cbuf);

    dim3 blk(32, 4);
    gemm_wmma_f32<<<dim3(BDIM / 16, (STDIM  + 255) / 256), blk, 0, stream>>>(cbuf, W_i2s, b_i2s, zrow, out_st, STDIM, STDIM, 1);
    gemm_wmma_f32<<<dim3(BDIM / 16, (OUTDIM + 255) / 256), blk, 0, stream>>>(cbuf, W_i2o, b_i2o, zrow, out_y, OUTDIM, OUTDIM, 1);
    gemm_wmma_f32<<<dim3(BDIM / 16, (UPD    + 255) / 256), blk, 0, stream>>>(cbuf, W_i2u, b_i2u, zrow, udb, UPD, UPD, 0);

    ntm_update_kernel<<<BDIM, 256, 0, stream>>>(wtg, mem, udb, out_mem, out_wt);
}